// MessagePassing_20822001451734
// MI455X (gfx1250) — compile-verified
//
#include <hip/hip_runtime.h>
#include <stdint.h>

#define D_FEAT 128
#define TILE   256          // edges staged per block iteration
#define BLOCK  256          // 8 waves (wave32)

// ---------------------------------------------------------------------------
// Zero-fill: must run every launch (atomics accumulate; harness replays graph)
// ---------------------------------------------------------------------------
__global__ void mp_zero_f4(float4* __restrict__ out, int n4) {
    int i = blockIdx.x * blockDim.x + threadIdx.x;
    int stride = gridDim.x * blockDim.x;
    float4 z = make_float4(0.f, 0.f, 0.f, 0.f);
    for (; i < n4; i += stride) out[i] = z;
}

// ---------------------------------------------------------------------------
// Scatter kernel:
//   - blocks grid-stride over tiles of 256 edges
//   - edge metadata (index_i, index_j, values) double-buffered into LDS via
//     CDNA5 async-to-LDS DMA (global_load_async_to_lds_b32 / s_wait_asynccnt)
//   - wave-per-edge: 32 lanes x float4 = 128 features (one b128 gather/lane)
//   - output accumulated with hardware non-returning f32 global atomics in
//     saddr form: uniform out-base in SGPR pair + 32-bit per-lane offset
// ---------------------------------------------------------------------------
__global__ __launch_bounds__(BLOCK) void mp_scatter(
    const float* __restrict__ x,
    const float* __restrict__ vals,
    const int*   __restrict__ idx_i,
    const int*   __restrict__ idx_j,
    float*       __restrict__ out,
    int E, int ntiles)
{
    __shared__ int   s_i[2][TILE];
    __shared__ int   s_j[2][TILE];
    __shared__ float s_v[2][TILE];

    const int tid  = threadIdx.x;
    const int wave = tid >> 5;
    const int lane = tid & 31;

    const uint64_t out_base = (uint64_t)(uintptr_t)out;   // stays in SGPRs

    // Async-DMA one tile's edge metadata into the chosen LDS buffer.
    auto prefetch = [&](int tile, int buf) {
        int e = tile * TILE + tid;
        if (e > E - 1) e = E - 1;                      // clamp tail (no OOB)
        uint64_t gi = (uint64_t)(uintptr_t)(idx_i + e);
        uint64_t gj = (uint64_t)(uintptr_t)(idx_j + e);
        uint64_t gv = (uint64_t)(uintptr_t)(vals  + e);
        // low 32 bits of a generic pointer to LDS == wave-relative LDS byte addr
        uint32_t li = (uint32_t)(uintptr_t)&s_i[buf][tid];
        uint32_t lj = (uint32_t)(uintptr_t)&s_j[buf][tid];
        uint32_t lv = (uint32_t)(uintptr_t)&s_v[buf][tid];
        asm volatile("global_load_async_to_lds_b32 %0, %1, off"
                     :: "v"(li), "v"(gi) : "memory");
        asm volatile("global_load_async_to_lds_b32 %0, %1, off"
                     :: "v"(lj), "v"(gj) : "memory");
        asm volatile("global_load_async_to_lds_b32 %0, %1, off"
                     :: "v"(lv), "v"(gv) : "memory");
    };

    int t   = blockIdx.x;
    int buf = 0;
    if (t < ntiles) prefetch(t, 0);

    for (; t < ntiles; t += gridDim.x) {
        // wait for this tile's async DMA, make it visible block-wide
        asm volatile("s_wait_asynccnt 0x0" ::: "memory");
        __syncthreads();

        // kick off next tile into the other buffer while we compute
        int tn = t + gridDim.x;
        if (tn < ntiles) prefetch(tn, buf ^ 1);

        // each wave processes 32 edges of this tile
        const int ebase = t * TILE + wave * 32;
        const int kmax  = (E - ebase) < 32 ? (E - ebase) : 32;
        for (int k = 0; k < kmax; ++k) {
            const int   slot = wave * 32 + k;
            const int   j = s_j[buf][slot];     // LDS broadcast reads
            const int   i = s_i[buf][slot];
            const float v = s_v[buf][slot];

            // gather one 512B row: 32 lanes x 16B = global_load_b128 each
            const float4* __restrict__ row =
                (const float4*)(x + (size_t)j * D_FEAT);
            float4 f = row[lane];

            const float a0 = v * f.x;
            const float a1 = v * f.y;
            const float a2 = v * f.z;
            const float a3 = v * f.w;

            // byte offset of out[i][lane*4]: i*512 + lane*16  (< 2^31)
            const uint32_t voff = ((uint32_t)i << 9) | ((uint32_t)lane << 4);

            // hardware f32 atomic adds, non-returning, saddr addressing mode
            asm volatile("global_atomic_add_f32 %0, %1, %2"
                         :: "v"(voff), "v"(a0), "s"(out_base) : "memory");
            asm volatile("global_atomic_add_f32 %0, %1, %2 offset:4"
                         :: "v"(voff), "v"(a1), "s"(out_base) : "memory");
            asm volatile("global_atomic_add_f32 %0, %1, %2 offset:8"
                         :: "v"(voff), "v"(a2), "s"(out_base) : "memory");
            asm volatile("global_atomic_add_f32 %0, %1, %2 offset:12"
                         :: "v"(voff), "v"(a3), "s"(out_base) : "memory");
        }

        // everyone done reading buf before it can be overwritten
        __syncthreads();
        buf ^= 1;
    }
}

// ---------------------------------------------------------------------------
// Host entry
// ---------------------------------------------------------------------------
extern "C" void kernel_launch(void* const* d_in, const int* in_sizes, int n_in,
                              void* d_out, int out_size, void* d_ws, size_t ws_size,
                              hipStream_t stream) {
    const float* x     = (const float*)d_in[0];   // [N, 128] f32
    const float* vals  = (const float*)d_in[1];   // [E] f32
    const int*   idx_i = (const int*)d_in[2];     // [E] i32 (targets)
    const int*   idx_j = (const int*)d_in[3];     // [E] i32 (sources)
    float*       out   = (float*)d_out;           // [N, 128] f32

    const int E  = in_sizes[1];
    const int n4 = out_size / 4;                  // out_size divisible by 4

    mp_zero_f4<<<dim3(2048), dim3(BLOCK), 0, stream>>>((float4*)out, n4);

    const int ntiles = (E + TILE - 1) / TILE;
    int blocks = ntiles < 1250 ? ntiles : 1250;   // 2500 tiles -> 2 tiles/block
    mp_scatter<<<dim3(blocks), dim3(BLOCK), 0, stream>>>(
        x, vals, idx_i, idx_j, out, E, ntiles);
}